// RejectionSampler_24206435680948
// MI455X (gfx1250) — compile-verified
//
#include <hip/hip_runtime.h>
#include <stdint.h>

// ---------------------------------------------------------------------------
// Rejection sampler (speculative decoding accept mask) for MI455X / gfx1250.
//
// Roofline: argmax over 2048 rows x 128000 fp32 = 1.05 GB streamed once.
// At 23.3 TB/s HBM that's ~45 us; this is a pure streaming reduction (no
// matmul -> WMMA inapplicable). CDNA5 path used: async global->LDS copies
// (GLOBAL_LOAD_ASYNC_TO_LDS_B128, ASYNCcnt). ASYNCcnt counts one per
// wave-wide instruction (512 B/wave), 6-bit deep, so an 8-deep LDS pipeline
// keeps 4 KB/wave in flight: ~8K resident waves x 4 KB = ~32 MB outstanding,
// enough to cover HBM latency at 23.3 TB/s.
// ---------------------------------------------------------------------------

#define THREADS 256
#define CHUNK   (THREADS * 4)   // 1024 floats = 4 KB per chunk
#define DEPTH   8               // pipeline depth (power of 2); 32 KB LDS

// Per-lane 16-byte async copy: VDST = LDS byte address, VADDR = 64-bit global
// address, GV mode (saddr = off). One instruction per wave -> ASYNCcnt += 1.
__device__ __forceinline__ void async_cp16(uint32_t lds_byte_off, const float* gp) {
  asm volatile("global_load_async_to_lds_b128 %0, %1, off"
               :: "v"(lds_byte_off), "v"(gp)
               : "memory");
}

// Async loads complete in order per wave, so after issuing up to chunk
// it+DEPTH-1, ASYNCcnt <= DEPTH-1 implies chunk `it` has fully landed in LDS.
// asm + "memory" clobber so the compiler cannot hoist ds_reads above the wait.
#define WAIT_ASYNC_STEADY() asm volatile("s_wait_asynccnt 7" ::: "memory")
#define WAIT_ASYNC_DRAIN()  asm volatile("s_wait_asynccnt 0" ::: "memory")

// Monotone key: larger float => larger key; equal floats => smaller index wins
// (matches jnp.argmax first-occurrence tie-break).
__device__ __forceinline__ unsigned long long pack_key(float v, unsigned idx) {
  unsigned u = __float_as_uint(v);
  u = (u & 0x80000000u) ? ~u : (u | 0x80000000u);
  return (((unsigned long long)u) << 32) |
         (unsigned long long)(0xFFFFFFFFu - idx);
}

__global__ __launch_bounds__(THREADS)
void rowargmax_kernel(const float* __restrict__ logits,
                      int* __restrict__ out_tok, int V) {
  __shared__ float buf[DEPTH][CHUNK];             // 32 KB ring of chunks
  __shared__ unsigned long long red[THREADS];     // 2 KB reduction tree

  const int tid = threadIdx.x;
  const int row = blockIdx.x;
  const float* rowp = logits + (size_t)row * (size_t)V;

  float    bv = -__builtin_inff();
  unsigned bi = 0u;

  const int nfull = V / CHUNK;  // 125 for V=128000

  // LDS byte offset of this thread's private 16B slot in buffer 0 (generic
  // pointer to __shared__ carries the LDS offset in its low 32 bits).
  const uint32_t lds_base = (uint32_t)(size_t)(&buf[0][tid * 4]);

  // --- prologue: fill the pipeline -----------------------------------------
  const int P = (nfull < DEPTH) ? nfull : DEPTH;
  for (int d = 0; d < P; ++d)
    async_cp16(lds_base + (uint32_t)(d * CHUNK * 4),
               rowp + (size_t)d * CHUNK + (size_t)tid * 4);

  // --- steady state: wait-for-oldest, consume, refill same slot ------------
  const int main_iters = nfull - P;   // nfull - DEPTH when pipeline is full
  for (int it = 0; it < main_iters; ++it) {
    WAIT_ASYNC_STEADY();              // chunk `it` (oldest) has landed
    const int slot = it & (DEPTH - 1);
    const float4 v4 = *(const float4*)&buf[slot][tid * 4];
    const unsigned base = (unsigned)(it * CHUNK + tid * 4);
    if (v4.x > bv) { bv = v4.x; bi = base;     }
    if (v4.y > bv) { bv = v4.y; bi = base + 1; }
    if (v4.z > bv) { bv = v4.z; bi = base + 2; }
    if (v4.w > bv) { bv = v4.w; bi = base + 3; }
    // Refill the slot just consumed with chunk it+DEPTH. Its global data
    // arrives an HBM round-trip later, long after our ds_read completed.
    async_cp16(lds_base + (uint32_t)(slot * CHUNK * 4),
               rowp + (size_t)(it + DEPTH) * CHUNK + (size_t)tid * 4);
  }

  // --- epilogue: drain the remaining <= DEPTH chunks -----------------------
  WAIT_ASYNC_DRAIN();
  for (int it = main_iters; it < nfull; ++it) {
    const int slot = it & (DEPTH - 1);
    const float4 v4 = *(const float4*)&buf[slot][tid * 4];
    const unsigned base = (unsigned)(it * CHUNK + tid * 4);
    if (v4.x > bv) { bv = v4.x; bi = base;     }
    if (v4.y > bv) { bv = v4.y; bi = base + 1; }
    if (v4.z > bv) { bv = v4.z; bi = base + 2; }
    if (v4.w > bv) { bv = v4.w; bi = base + 3; }
  }

  // Tail for V not a multiple of CHUNK (not taken for V=128000).
  for (int g = nfull * CHUNK + tid; g < V; g += THREADS) {
    const float v = rowp[g];
    if (v > bv) { bv = v; bi = (unsigned)g; }
  }

  // Cross-thread argmax via packed-key max tree in LDS.
  red[tid] = pack_key(bv, bi);
  __syncthreads();
  #pragma unroll
  for (int s = THREADS / 2; s > 0; s >>= 1) {
    if (tid < s) {
      const unsigned long long a = red[tid], b = red[tid + s];
      red[tid] = (b > a) ? b : a;
    }
    __syncthreads();
  }
  if (tid == 0)
    out_tok[row] = (int)(0xFFFFFFFFu - (uint32_t)red[0]);
}

// Accept-mask + bonus token. m = number of leading spec tokens that match the
// greedy argmax (0..K). Emit positions 0..m (accepted prefix + bonus /
// correction token), INVALID_TOKEN_ID (-1) elsewhere.
__global__ void rejection_kernel(const int* __restrict__ toks,
                                 const int* __restrict__ spec,
                                 float* __restrict__ out, int B, int K) {
  const int b = blockIdx.x * blockDim.x + threadIdx.x;
  if (b >= B) return;
  const int S = K + 1;
  int m = 0;
  for (int j = 0; j < K; ++j) {
    if (m == j && toks[b * S + j] == spec[b * K + j]) m = j + 1;
  }
  for (int j = 0; j < S; ++j) {
    const int t = toks[b * S + j];
    out[b * S + j] = (j <= m) ? (float)t : -1.0f;
  }
}

extern "C" void kernel_launch(void* const* d_in, const int* in_sizes, int n_in,
                              void* d_out, int out_size, void* d_ws, size_t ws_size,
                              hipStream_t stream) {
  const float* logits = (const float*)d_in[0];          // (B*(K+1), V) fp32
  const int*   spec   = (const int*)d_in[1];            // (B, K) int32

  const int rows = out_size;                // B*(K+1) = 2048
  const int BK   = in_sizes[1];             // B*K     = 1792
  const int B    = rows - BK;               // 256
  const int K    = BK / B;                  // 7
  const int V    = (int)((long long)in_sizes[0] / (long long)rows); // 128000

  int* argm = (int*)d_ws;                   // rows * 4 bytes of scratch

  rowargmax_kernel<<<rows, THREADS, 0, stream>>>(logits, argm, V);
  rejection_kernel<<<(B + 255) / 256, 256, 0, stream>>>(argm, spec,
                                                        (float*)d_out, B, K);
}